// AncProbsLayer_26388279067093
// MI455X (gfx1250) — compile-verified
//
#include <hip/hip_runtime.h>

// Problem constants (match reference)
#define S20 20
#define Bc 1024
#define Lc 512
#define Hc 2
#define Kc 2
#define NRATES 100000

typedef __attribute__((ext_vector_type(2))) float v2f;
typedef __attribute__((ext_vector_type(8))) float v8f;

__device__ __forceinline__ float softplus_f(float x) {
  // numerically stable softplus, matches jax.nn.softplus
  return fmaxf(x, 0.f) + log1pf(expf(-fabsf(x)));
}

// Round-robin ("chess tournament") schedule: round r, pair i in [0,10)
// produces disjoint (p,q) pairs covering all 20 indices.
__device__ __forceinline__ void jacobi_pair(int round, int i, int& p, int& q) {
  int a = (i == 0) ? 0 : (1 + ((i - 1 + round) % 19));
  int j = 19 - i;                       // j in [10,19], always >= 1
  int b = 1 + ((j - 1 + round) % 19);
  p = (a < b) ? a : b;
  q = (a < b) ? b : a;
}

// ---------------------------------------------------------------------------
// Kernel 1: per (h,k) build R, p, Q, Sm and eigendecompose Sm with parallel
// cyclic Jacobi. Outputs lam (HK*20), U (HK*400), sqrt_p, inv_sqrt_p.
// ---------------------------------------------------------------------------
__global__ void __launch_bounds__(256)
gtr_precompute(const float* __restrict__ exch, const float* __restrict__ eq,
               float* __restrict__ olam, float* __restrict__ oU,
               float* __restrict__ osp, float* __restrict__ oisp)
{
  __shared__ float sR[S20][S20];
  __shared__ float sQ[S20][S20];
  __shared__ float sA[S20][S20];
  __shared__ float sV[S20][S20];
  __shared__ float spv[S20];
  __shared__ float ssq[S20];
  __shared__ float sisq[S20];
  __shared__ float srow[S20];
  __shared__ float scs[10][2];

  const int hk   = blockIdx.x;          // h*K + k
  const int tid  = threadIdx.x;
  const int mb   = hk * 400;
  const int vb   = hk * 20;

  // R = softplus(0.5*(ker + ker^T)) with zero diagonal
  for (int t = tid; t < 400; t += 256) {
    int i = t / 20, j = t % 20;
    float x = 0.5f * (exch[mb + i * 20 + j] + exch[mb + j * 20 + i]);
    sR[i][j] = (i == j) ? 0.f : softplus_f(x);
  }
  // p = softmax(eq row); sqrt_p, 1/sqrt_p
  if (tid < 20) {
    float mx = -1e30f;
    for (int j = 0; j < 20; ++j) mx = fmaxf(mx, eq[vb + j]);
    float sum = 0.f;
    for (int j = 0; j < 20; ++j) sum += expf(eq[vb + j] - mx);
    float p = expf(eq[vb + tid] - mx) / sum;
    spv[tid]  = p;
    float sq  = sqrtf(p);
    ssq[tid]  = sq;
    sisq[tid] = 1.f / sq;
  }
  __syncthreads();

  // Q = R * p_j, diagonal = -rowsum
  if (tid < 20) {
    float rs = 0.f;
    for (int j = 0; j < 20; ++j) {
      float qv = sR[tid][j] * spv[j];
      sQ[tid][j] = qv;
      rs += qv;
    }
    sQ[tid][tid] = -rs;                 // R_ii == 0 so Q_ii was 0
    srow[tid] = rs;
  }
  __syncthreads();

  // normalize by expected mutation rate; build symmetric Sm; V = I
  float mue = 0.f;
  for (int j = 0; j < 20; ++j) mue += spv[j] * srow[j];
  const float inv_mue = 1.f / fmaxf(mue, 1e-16f);
  for (int t = tid; t < 400; t += 256) {
    int i = t / 20, j = t % 20;
    float a  = ssq[i] * sQ[i][j] * sisq[j];
    float b2 = ssq[j] * sQ[j][i] * sisq[i];
    sA[i][j] = 0.5f * (a + b2) * inv_mue;
    sV[i][j] = (i == j) ? 1.f : 0.f;
  }
  __syncthreads();

  // Parallel cyclic Jacobi: 12 sweeps x 19 rounds x 10 disjoint rotations
  for (int sweep = 0; sweep < 12; ++sweep) {
    for (int round = 0; round < 19; ++round) {
      if (tid < 10) {
        int p, q; jacobi_pair(round, tid, p, q);
        float app = sA[p][p], aqq = sA[q][q], apq = sA[p][q];
        float c = 1.f, s = 0.f;
        if (fabsf(apq) > 1e-30f) {
          float th = (aqq - app) / (2.f * apq);
          float tt = 1.f / (fabsf(th) + sqrtf(th * th + 1.f));
          if (th < 0.f) tt = -tt;
          c = 1.f / sqrtf(tt * tt + 1.f);
          s = tt * c;
        }
        scs[tid][0] = c; scs[tid][1] = s;
      }
      __syncthreads();
      // row phase: B = G^T A  (pairs own disjoint rows)
      if (tid < 200) {
        int i = tid / 20, j = tid % 20;
        int p, q; jacobi_pair(round, i, p, q);
        float c = scs[i][0], s = scs[i][1];
        float ap = sA[p][j], aq = sA[q][j];
        sA[p][j] = c * ap - s * aq;
        sA[q][j] = s * ap + c * aq;
      }
      __syncthreads();
      // column phase: A = B G ; accumulate V = V G
      if (tid < 200) {
        int i = tid / 20, j = tid % 20;
        int p, q; jacobi_pair(round, i, p, q);
        float c = scs[i][0], s = scs[i][1];
        float ap = sA[j][p], aq = sA[j][q];
        sA[j][p] = c * ap - s * aq;
        sA[j][q] = s * ap + c * aq;
        float vp = sV[j][p], vq = sV[j][q];
        sV[j][p] = c * vp - s * vq;
        sV[j][q] = s * vp + c * vq;
      }
      __syncthreads();
    }
  }

  if (tid < 20) {
    olam[vb + tid] = sA[tid][tid];
    osp[vb + tid]  = ssq[tid];
    oisp[vb + tid] = sisq[tid];
  }
  for (int t = tid; t < 400; t += 256)
    oU[mb + t] = sV[t / 20][t % 20];    // columns of V are eigenvectors
}

// ---------------------------------------------------------------------------
// Kernel 2: one block per (b,h). Build P_k = D^-1 U diag(e^{tau lam}) U^T D
// in LDS, then WMMA the (512 x 20) one-hot slab against both 20x20 P's.
// ---------------------------------------------------------------------------
__global__ void __launch_bounds__(256)
ancprobs_main(const float* __restrict__ in, const int* __restrict__ rate_idx,
              const float* __restrict__ tau_kernel,
              const float* __restrict__ wlam, const float* __restrict__ wU,
              const float* __restrict__ wsp, const float* __restrict__ wisp,
              float* __restrict__ out)
{
  __shared__ float sU[Kc][S20][S20];
  __shared__ float sE[Kc][S20];
  __shared__ float sSp[Kc][S20];
  __shared__ float sIsp[Kc][S20];
  __shared__ float sP[Kc][S20][S20];

  const int bx  = blockIdx.x;
  const int b   = bx >> 1;
  const int h   = bx & 1;
  const int tid = threadIdx.x;

  // tau[b,h] (uniform across the block -> scalarized by compiler)
  const int   ridx = rate_idx[b * Hc + h];
  const float tau  = softplus_f(tau_kernel[h * NRATES + ridx]);

  for (int t = tid; t < Kc * 400; t += 256) {
    int k = t / 400, r = t % 400;
    sU[k][r / 20][r % 20] = wU[(h * Kc + k) * 400 + r];
  }
  for (int t = tid; t < Kc * 20; t += 256) {
    int k = t / 20, s = t % 20;
    sE[k][s]   = expf(tau * wlam[(h * Kc + k) * 20 + s]);
    sSp[k][s]  = wsp[(h * Kc + k) * 20 + s];
    sIsp[k][s] = wisp[(h * Kc + k) * 20 + s];
  }
  __syncthreads();

  // P[k][i][j] = isp_i * (sum_s U_is E_s U_js) * sp_j
  for (int t = tid; t < Kc * 400; t += 256) {
    int k = t / 400, r = t % 400, i = r / 20, j = r % 20;
    float acc = 0.f;
#pragma unroll
    for (int s = 0; s < 20; ++s) acc += sU[k][i][s] * sE[k][s] * sU[k][j][s];
    sP[k][i][j] = sIsp[k][i] * acc * sSp[k][j];
  }
  __syncthreads();

  const int lid  = tid & 31;
  const int wv   = tid >> 5;            // 8 waves
  const int half = lid >> 4;            // 0: K0/K1 regs, 1: K2/K3 regs
  const int l16  = lid & 15;

  // B-fragments (loop-invariant): B rows = z (contraction), cols = output s.
  // Layout per ISA 7.12.2 (mirror of A): lane<16 -> {K0,K1}, lane>=16 -> {K2,K3}.
  v2f bf[Kc][2][5];
#pragma unroll
  for (int k = 0; k < Kc; ++k)
#pragma unroll
    for (int t = 0; t < 2; ++t)
#pragma unroll
      for (int c = 0; c < 5; ++c) {
        int n  = t * 16 + l16;
        int z0 = 4 * c + 2 * half;
        float b0 = (n < 20) ? sP[k][z0][n]     : 0.f;
        float b1 = (n < 20) ? sP[k][z0 + 1][n] : 0.f;
        v2f bv = {b0, b1};
        bf[k][t][c] = bv;
      }

  const size_t inBase  = (size_t)b * Lc * (Hc * S20) + (size_t)h * S20;
  const size_t outBase = (size_t)b * Lc * (Hc * Kc * S20) + (size_t)h * (Kc * S20);

  for (int it = 0; it < 4; ++it) {      // 32 row-tiles / 8 waves
    const int l0   = (wv + 8 * it) * 16;
    const int arow = l0 + l16;

    // A-fragments: 16 l-rows x 4 z per chunk; aligned 8B loads (z0 even)
    v2f af[5];
#pragma unroll
    for (int c = 0; c < 5; ++c) {
      int z = 4 * c + 2 * half;
      af[c] = *reinterpret_cast<const v2f*>(in + inBase + (size_t)arow * (Hc * S20) + z);
    }

    v8f acc[Kc][2];
#pragma unroll
    for (int k = 0; k < Kc; ++k)
#pragma unroll
      for (int t = 0; t < 2; ++t) {
        v8f a = {0.f, 0.f, 0.f, 0.f, 0.f, 0.f, 0.f, 0.f};
#pragma unroll
        for (int c = 0; c < 5; ++c) {
          a = __builtin_amdgcn_wmma_f32_16x16x4_f32(
              /*neg_a=*/false, af[c], /*neg_b=*/false, bf[k][t][c],
              /*c_mod=*/(short)0, a, /*reuse_a=*/false, /*reuse_b=*/false);
        }
        acc[k][t] = a;
      }

    // D layout: VGPR r -> (M=r, N=lane) lanes 0-15 ; (M=8+r, N=lane-16) lanes 16-31
#pragma unroll
    for (int k = 0; k < Kc; ++k)
#pragma unroll
      for (int t = 0; t < 2; ++t) {
        int n = t * 16 + l16;
        if (n < 20) {
#pragma unroll
          for (int r = 0; r < 8; ++r) {
            int l = l0 + half * 8 + r;
            out[outBase + (size_t)l * (Hc * Kc * S20) + k * S20 + n] = acc[k][t][r];
          }
        }
      }
  }
}

extern "C" void kernel_launch(void* const* d_in, const int* in_sizes, int n_in,
                              void* d_out, int out_size, void* d_ws, size_t ws_size,
                              hipStream_t stream) {
  const float* inputs       = (const float*)d_in[0];
  const int*   rate_indices = (const int*)d_in[1];
  const float* tau_kernelp  = (const float*)d_in[2];
  const float* exch         = (const float*)d_in[3];
  const float* eq           = (const float*)d_in[4];
  float* out = (float*)d_out;

  float* ws    = (float*)d_ws;
  float* wlam  = ws;            // H*K*20   = 80
  float* wU    = ws + 80;       // H*K*400  = 1600
  float* wsp   = ws + 1680;     // 80
  float* wisp  = ws + 1760;     // 80

  gtr_precompute<<<Hc * Kc, 256, 0, stream>>>(exch, eq, wlam, wU, wsp, wisp);
  ancprobs_main<<<Bc * Hc, 256, 0, stream>>>(inputs, rate_indices, tau_kernelp,
                                             wlam, wU, wsp, wisp, out);
}